// DiagDot_27917287424359
// MI455X (gfx1250) — compile-verified
//
#include <hip/hip_runtime.h>

// CDNA5 / gfx1250. Bandwidth-bound weighted-sum-of-squares row reduction.
// out[r] = sum_j x[r][j]^2 * w[j]
//
// Design (MI455X): 23.3 TB/s HBM floor ~11us for the 256MiB input stream.
// - one wave32 per row; lanes stream float4 (global_load_b128), coalesced.
// - weight (16KiB) staged once per workgroup into LDS (320KiB/WGP).
// - final 32-lane reduction done with one V_WMMA_F32_16X16X4_F32:
//     A[m][0]=p[m], A[m][2]=p[m+16] (VGPR0 = partial across all 32 lanes),
//     B = ones  =>  D[m][n] = p[m] + p[m+16].
//   Summing the 8 D VGPRs per lane + one v_readlane(16) yields the full
//   32-way fp32 sum. Exercises the CDNA5 matrix pipe at zero bandwidth cost.

typedef __attribute__((ext_vector_type(2))) float v2f;
typedef __attribute__((ext_vector_type(8))) float v8f;

#define FEATS 4096
#define ROWS  16384
#define BLOCK 256
#define WAVES_PER_BLOCK 8   // 256 / 32

__global__ __launch_bounds__(BLOCK)
void DiagDot_27917287424359_kernel(const float* __restrict__ inp,
                                   const float* __restrict__ w,
                                   float* __restrict__ out) {
    __shared__ float4 ws4[FEATS / 4];   // 16 KiB weight tile in LDS

    const int tid = threadIdx.x;

    // Stage weight into LDS: 1024 float4, 256 threads -> 4 b128 each.
    {
        const float4* wsrc = reinterpret_cast<const float4*>(w);
#pragma unroll
        for (int u = 0; u < 4; ++u) {
            const int idx = tid + BLOCK * u;
            ws4[idx] = wsrc[idx];
        }
    }
    __syncthreads();

    const int lane = tid & 31;
    const int wave = tid >> 5;
    const int row  = blockIdx.x * WAVES_PER_BLOCK + wave;
    if (row >= ROWS) return;   // wave-uniform (grid divides exactly)

    const float4* __restrict__ xrow =
        reinterpret_cast<const float4*>(inp + (size_t)row * FEATS);

    // 1024 float4 per row / 32 lanes = 32 per lane; unroll 8 to keep
    // 8 b128 loads in flight per lane.
    float acc = 0.0f;
#pragma unroll 8
    for (int j = lane; j < FEATS / 4; j += 32) {
        const float4 x  = xrow[j];
        const float4 wq = ws4[j];
        acc = fmaf(x.x * x.x, wq.x, acc);
        acc = fmaf(x.y * x.y, wq.y, acc);
        acc = fmaf(x.z * x.z, wq.z, acc);
        acc = fmaf(x.w * x.w, wq.w, acc);
    }

    // ---- WMMA cross-lane reduction (fp32 exact path) ----
    // A (16x4 f32, 2 VGPRs/lane): VGPR0 = per-lane partial, VGPR1 = 0.
    //   => A[m][0] = acc(lane m), A[m][2] = acc(lane m+16), A[m][1]=A[m][3]=0.
    // B (4x16 f32) = ones.  D[m][n] = acc(m) + acc(m+16) for every n.
    v2f a; a.x = acc;  a.y = 0.0f;
    v2f b; b.x = 1.0f; b.y = 1.0f;
    v8f c = {};
    v8f d = __builtin_amdgcn_wmma_f32_16x16x4_f32(
        /*neg_a=*/false, a, /*neg_b=*/false, b,
        /*c_mod=*/(short)0, c, /*reuse_a=*/false, /*reuse_b=*/false);

    // Lanes 0-15 hold D rows 0..7, lanes 16-31 hold rows 8..15:
    // summing the 8 VGPRs gives (lanes 0-15) sum of s_0..s_7 and
    // (lanes 16-31) sum of s_8..s_15, where s_m = acc(m)+acc(m+16).
    float t = 0.0f;
#pragma unroll
    for (int i = 0; i < 8; ++i) t += d[i];

    // Combine the two halves: lane0 total = sum over all 32 partials.
    const float hi = __builtin_bit_cast(
        float, __builtin_amdgcn_readlane(__builtin_bit_cast(int, t), 16));

    if (lane == 0) out[row] = t + hi;
}

extern "C" void kernel_launch(void* const* d_in, const int* in_sizes, int n_in,
                              void* d_out, int out_size, void* d_ws, size_t ws_size,
                              hipStream_t stream) {
    (void)in_sizes; (void)n_in; (void)d_ws; (void)ws_size; (void)out_size;
    const float* inp = (const float*)d_in[0];   // (16384, 4096) f32
    const float* w   = (const float*)d_in[1];   // (4096,) f32
    float* out       = (float*)d_out;           // (16384,) f32

    const int blocks = ROWS / WAVES_PER_BLOCK;  // 2048
    DiagDot_27917287424359_kernel<<<blocks, BLOCK, 0, stream>>>(inp, w, out);
}